// ManoLayer_55997783605550
// MI455X (gfx1250) — compile-verified
//
#include <hip/hip_runtime.h>
#include <math.h>

#define DEV __device__ __forceinline__

typedef __attribute__((ext_vector_type(2))) float v2f;
typedef __attribute__((ext_vector_type(8))) float v8f;

constexpr int BS    = 4096;
constexpr int NV    = 778;
constexpr int NVD   = NV * 3;      // 2334 (v,d flattened)
constexpr int NSTR  = 2336;        // padded column stride for B matrix
constexpr int KPAD  = 148;         // 135 pose_feat + 10 shape + 3 zero pad
constexpr int NT_N  = 146;         // ceil(2334/16) col tiles
constexpr int NT_M  = BS / 16;     // 256 batch tiles
constexpr int VTILE = 49;          // ceil(778/16) vertex tiles
constexpr int NVP   = VTILE * 16;  // 784 padded vertex rows

// workspace layout (float offsets)
constexpr size_t OFF_BPK  = 0;                                   // 148*2336 (packed)
constexpr size_t OFF_VTP  = OFF_BPK + (size_t)KPAD * NSTR;       // 2336
constexpr size_t OFF_JS   = OFF_VTP + NSTR;                      // 480
constexpr size_t OFF_JT   = OFF_JS + 480;                        // 48
constexpr size_t OFF_AXIS = OFF_JT + 48;                         // BS*48
constexpr size_t OFF_PF   = OFF_AXIS + (size_t)BS * 48;          // BS*148
constexpr size_t OFF_SE3  = OFF_PF + (size_t)BS * KPAD;          // BS*256 (packed)
constexpr size_t OFF_JL   = OFF_SE3 + (size_t)BS * 256;          // BS*48
constexpr size_t OFF_POSP = OFF_JL + (size_t)BS * 48;            // BS*48
constexpr size_t OFF_HCP  = OFF_POSP + (size_t)BS * 48;          // 2304 (packed)
constexpr size_t OFF_MNP  = OFF_HCP + 2304;                      // 48
constexpr size_t OFF_WP   = OFF_MNP + 48;                        // 784*16
// total ~2.61M floats ~= 10.4 MB

__device__ const int d_NEW_ORDER[21] = {0,13,14,15,16,1,2,3,17,4,5,6,18,10,11,12,19,7,8,9,20};
__device__ const int d_TIPS[5]       = {745,333,444,555,672};

// ---- fp32 WMMA 16x16x4: D = A(16x4) * B(4x16) + C --------------------------
DEV v8f wmma4(v2f a, v2f b, v8f c) {
  return __builtin_amdgcn_wmma_f32_16x16x4_f32(
      false, a, false, b, (short)0, c, false, false);
}

DEV v2f ld2(const float* p) { return *(const v2f*)p; }

// packed position of SE3 joint j, element e: B rows {4ks+2hi, 4ks+2hi+1} adjacent
DEV int se3base(int j) { return (((j >> 2) * 2 + ((j >> 1) & 1)) * 16) * 2 + (j & 1); }

// ---- Rodrigues (matches axis_to_rmat, eps 1e-12) ----------------------------
DEV void rodrigues(float ax, float ay, float az, float* R) {
  float a2  = ax * ax + ay * ay + az * az + 1e-12f;
  float ang = sqrtf(a2);
  float inv = 1.0f / ang;
  float ux = ax * inv, uy = ay * inv, uz = az * inv;
  float c = cosf(ang), s = sinf(ang);
  float oc = 1.0f - c;
  R[0] = c + oc * ux * ux;      R[1] = oc * ux * uy - s * uz; R[2] = oc * ux * uz + s * uy;
  R[3] = oc * uy * ux + s * uz; R[4] = c + oc * uy * uy;      R[5] = oc * uy * uz - s * ux;
  R[6] = oc * uz * ux - s * uy; R[7] = oc * uz * uy + s * ux; R[8] = c + oc * uz * uz;
}

// j_tpose[i] = Jt[i] + JS[i] @ shape   (JS precomputed, batch independent)
DEV void jtpose(int i, const float* sh, const float* __restrict__ JS,
                const float* __restrict__ Jt, float* out) {
#pragma unroll
  for (int d = 0; d < 3; ++d) {
    float acc = Jt[i * 3 + d];
#pragma unroll
    for (int s = 0; s < 10; ++s) acc += JS[(i * 3 + d) * 10 + s] * sh[s];
    out[d] = acc;
  }
}

// ============================================================================
// K0a: build packed combined dirs matrix: Bpk[((2ks+hi)*NSTR + vd)*2 + p]
//      = [posedirs | shapedirs]^T row (4ks+2hi+p), col vd.  Also padded v_template.
// ============================================================================
__global__ __launch_bounds__(256) void k0_bpk(
    const float* __restrict__ posedirs, const float* __restrict__ shapedirs,
    const float* __restrict__ vtemp, float* __restrict__ Bpk,
    float* __restrict__ vtempP) {
  int idx = blockIdx.x * 256 + threadIdx.x;
  if (idx < KPAD * NSTR) {
    int p = idx & 1;
    int t = idx >> 1;
    int vd = t % NSTR, q = t / NSTR;
    int k = (q >> 1) * 4 + (q & 1) * 2 + p;
    float val = 0.0f;
    if (vd < NVD) {
      int v = vd / 3, d = vd % 3;
      if (k < 135)      val = posedirs[v * 405 + d * 135 + k];
      else if (k < 145) val = shapedirs[v * 30 + d * 10 + (k - 135)];
    }
    Bpk[idx] = val;
  }
  if (idx < NSTR) vtempP[idx] = (idx < NVD) ? vtemp[idx] : 0.0f;
}

// ============================================================================
// K0b: JS[j,d,s] = sum_v J[j,v]*shapedirs[v,d,s];  Jt[j,d] = sum_v J[j,v]*vt[v,d]
// ============================================================================
__global__ __launch_bounds__(256) void k0_js(
    const float* __restrict__ Jreg, const float* __restrict__ shapedirs,
    const float* __restrict__ vtemp, float* __restrict__ JS, float* __restrict__ Jt) {
  int idx = blockIdx.x * 256 + threadIdx.x;
  if (idx < 480) {
    int j = idx / 30, rem = idx % 30, d = rem / 10, s = rem % 10;
    float acc = 0.0f;
    for (int v = 0; v < NV; ++v) acc += Jreg[j * NV + v] * shapedirs[v * 30 + d * 10 + s];
    JS[idx] = acc;
  } else if (idx < 528) {
    int q = idx - 480, j = q / 3, d = q % 3;
    float acc = 0.0f;
    for (int v = 0; v < NV; ++v) acc += Jreg[j * NV + v] * vtemp[v * 3 + d];
    Jt[q] = acc;
  }
}

// ============================================================================
// K0c: zero-pad pose/mean/weights and pack HC so the WMMA loops are branch-free
// ============================================================================
__global__ __launch_bounds__(256) void k0_pack(
    const float* __restrict__ pose, const float* __restrict__ HC,
    const float* __restrict__ mean, const float* __restrict__ weights,
    float* __restrict__ posP, float* __restrict__ HCP,
    float* __restrict__ meanP, float* __restrict__ WP) {
  int idx = blockIdx.x * 256 + threadIdx.x;
  if (idx < BS * 48) {                              // padded pose rows
    int j = idx % 48, b = idx / 48;
    posP[idx] = (j < 45) ? pose[b * 45 + j] : 0.0f;
    return;
  }
  idx -= BS * 48;
  if (idx < 2304) {                                 // packed padded HC (48x48)
    int p = idx & 1;
    int t = idx >> 1;
    int n = t % 48, q = t / 48;
    int k = (q >> 1) * 4 + (q & 1) * 2 + p;
    HCP[idx] = (k < 45 && n < 45) ? HC[k * 45 + n] : 0.0f;
    return;
  }
  idx -= 2304;
  if (idx < 48) { meanP[idx] = (idx < 45) ? mean[idx] : 0.0f; return; }
  idx -= 48;
  if (idx < NVP * 16) {                             // padded weights rows
    int v = idx >> 4;
    WP[idx] = (v < NV) ? weights[idx] : 0.0f;
  }
}

// ============================================================================
// K_axis: axis = pose @ hands_components + mean   (branch-free WMMA fp32)
// ============================================================================
__global__ __launch_bounds__(256) void k_axis(
    const float* __restrict__ posP, const float* __restrict__ HCP,
    const float* __restrict__ meanP, float* __restrict__ axisW) {
  int wave = (blockIdx.x * 256 + threadIdx.x) >> 5;
  int lane = threadIdx.x & 31;
  int tileM = wave / 3, tileN = wave % 3;
  int m0 = tileM * 16, n0 = tileN * 16;
  int row = lane & 15, hi = lane >> 4, col = row;
  int n = n0 + col;
  const float* pr = posP + (size_t)(m0 + row) * 48;
  v8f acc = {0, 0, 0, 0, 0, 0, 0, 0};
#pragma unroll
  for (int ks = 0; ks < 12; ++ks) {
    v2f a = ld2(pr + ks * 4 + 2 * hi);
    v2f b = ld2(HCP + ((ks * 2 + hi) * 48 + n) * 2);
    acc = wmma4(a, b, acc);
  }
  float mn = meanP[n];
#pragma unroll
  for (int i = 0; i < 8; ++i)
    axisW[(size_t)(m0 + i + 8 * hi) * 48 + n] = acc[i] + mn;
}

// ============================================================================
// K1: per-batch prep (1 thread per batch element):
//   pose_feat -> PF, joints via JS/Jt, SE3 chain -> SE3 packed (WMMA-B order), jlist
// ============================================================================
__global__ __launch_bounds__(256) void k1_prep(
    const float* __restrict__ rootR, const float* __restrict__ shape,
    const float* __restrict__ axisW, const float* __restrict__ JS,
    const float* __restrict__ Jt, float* __restrict__ PF,
    float* __restrict__ SE3pk, float* __restrict__ jl) {
  int b = blockIdx.x * 256 + threadIdx.x;
  if (b >= BS) return;
  float sh[10];
#pragma unroll
  for (int s = 0; s < 10; ++s) sh[s] = shape[b * 10 + s];
  const float* ax = axisW + (size_t)b * 48;
  float* pf = PF + (size_t)b * KPAD;
  float* Sb = SE3pk + (size_t)b * 256;

  // pose_feat = rot - I (row-major, joint-major) | shape | zero pad
#pragma unroll
  for (int j = 0; j < 15; ++j) {
    float R[9];
    rodrigues(ax[3 * j], ax[3 * j + 1], ax[3 * j + 2], R);
#pragma unroll
    for (int r = 0; r < 3; ++r)
#pragma unroll
      for (int c = 0; c < 3; ++c)
        pf[9 * j + 3 * r + c] = R[3 * r + c] - (r == c ? 1.0f : 0.0f);
  }
#pragma unroll
  for (int s = 0; s < 10; ++s) pf[135 + s] = sh[s];
  pf[145] = 0.0f; pf[146] = 0.0f; pf[147] = 0.0f;

  // root SE3: R = root_rotation, t = j0 - R@j0
  float Rr[9];
#pragma unroll
  for (int k = 0; k < 9; ++k) Rr[k] = rootR[b * 9 + k];
  float j0[3]; jtpose(0, sh, JS, Jt, j0);
  float rt[3];
#pragma unroll
  for (int d = 0; d < 3; ++d)
    rt[d] = j0[d] - (Rr[3 * d] * j0[0] + Rr[3 * d + 1] * j0[1] + Rr[3 * d + 2] * j0[2]);
  {
    int base = se3base(0);
#pragma unroll
    for (int r = 0; r < 3; ++r) {
      Sb[base + 2 * (4 * r + 0)] = Rr[3 * r];
      Sb[base + 2 * (4 * r + 1)] = Rr[3 * r + 1];
      Sb[base + 2 * (4 * r + 2)] = Rr[3 * r + 2];
      Sb[base + 2 * (4 * r + 3)] = rt[r];
    }
    Sb[base + 24] = 0.0f; Sb[base + 26] = 0.0f; Sb[base + 28] = 0.0f; Sb[base + 30] = 1.0f;
  }
  float* jlb = jl + (size_t)b * 48;
  jlb[0] = j0[0]; jlb[1] = j0[1]; jlb[2] = j0[2];

  // 5 chains of 3 joints each hanging off the root
#pragma unroll
  for (int c = 0; c < 5; ++c) {
    float P[12];  // parent global SE3 (3x4)
#pragma unroll
    for (int r = 0; r < 3; ++r) {
      P[4 * r] = Rr[3 * r]; P[4 * r + 1] = Rr[3 * r + 1];
      P[4 * r + 2] = Rr[3 * r + 2]; P[4 * r + 3] = rt[r];
    }
#pragma unroll
    for (int s = 0; s < 3; ++s) {
      int i = 3 * c + s + 1;
      float R[9];
      rodrigues(ax[3 * (i - 1)], ax[3 * (i - 1) + 1], ax[3 * (i - 1) + 2], R);
      float ji[3]; jtpose(i, sh, JS, Jt, ji);
      // jlist[i] = P.R @ ji + P.t  (parent global transform)
#pragma unroll
      for (int d = 0; d < 3; ++d)
        jlb[3 * i + d] = P[4 * d] * ji[0] + P[4 * d + 1] * ji[1] + P[4 * d + 2] * ji[2] + P[4 * d + 3];
      // local t = ji - R@ji
      float ti[3];
#pragma unroll
      for (int d = 0; d < 3; ++d)
        ti[d] = ji[d] - (R[3 * d] * ji[0] + R[3 * d + 1] * ji[1] + R[3 * d + 2] * ji[2]);
      // compose: N = P * (R, ti)
      float Nw[12];
#pragma unroll
      for (int r = 0; r < 3; ++r) {
#pragma unroll
        for (int cc = 0; cc < 3; ++cc)
          Nw[4 * r + cc] = P[4 * r] * R[cc] + P[4 * r + 1] * R[3 + cc] + P[4 * r + 2] * R[6 + cc];
        Nw[4 * r + 3] = P[4 * r] * ti[0] + P[4 * r + 1] * ti[1] + P[4 * r + 2] * ti[2] + P[4 * r + 3];
      }
#pragma unroll
      for (int k = 0; k < 12; ++k) P[k] = Nw[k];
      int base = se3base(i);
#pragma unroll
      for (int e = 0; e < 12; ++e) Sb[base + 2 * e] = P[e];
      Sb[base + 24] = 0.0f; Sb[base + 26] = 0.0f; Sb[base + 28] = 0.0f; Sb[base + 30] = 1.0f;
    }
  }
}

// ============================================================================
// K2: v_tpose = v_template + [posedirs|shapedirs]^T @ [pose_feat|shape]
//   fp32 WMMA, K=148 (37 steps), branch-free b64 feeds. Output staged in d_out.
// ============================================================================
__global__ __launch_bounds__(256) void k2_vtpose(
    const float* __restrict__ Bpk, const float* __restrict__ PF,
    const float* __restrict__ vtempP, float* __restrict__ vout) {
  int wave = (blockIdx.x * 256 + threadIdx.x) >> 5;
  int lane = threadIdx.x & 31;
  int tileN = wave % NT_N;
  int tileM = wave / NT_N;
  int m0 = tileM * 16, n0 = tileN * 16;
  int row = lane & 15, hi = lane >> 4, col = row;
  int vd = n0 + col;
  float vt = vtempP[vd];                      // C init: v_template per column
  v8f acc = {vt, vt, vt, vt, vt, vt, vt, vt};
  const float* pfrow = PF + (size_t)(m0 + row) * KPAD;
  const float* bcol  = Bpk + (size_t)vd * 2 + (size_t)hi * NSTR * 2;
#pragma unroll
  for (int ks = 0; ks < KPAD / 4; ++ks) {
    v2f a = ld2(pfrow + ks * 4 + 2 * hi);
    v2f b = ld2(bcol + (size_t)ks * (2 * NSTR * 2));
    acc = wmma4(a, b, acc);
  }
  if (vd < NVD) {
#pragma unroll
    for (int i = 0; i < 8; ++i)
      vout[(size_t)(m0 + i + 8 * hi) * NVD + vd] = acc[i];
  }
}

// ============================================================================
// K3: skinning. Per wave: one (batch, 16-vertex) tile.
//   SE3_v = weights @ SE3 (WMMA, K=16 joints, N=16 flattened 4x4), LDS transpose,
//   then v_out = SE3_v[:3,:3]@v_tpose + SE3_v[:3,3] - center + trans  (in place)
// ============================================================================
__global__ __launch_bounds__(256) void k3_skin(
    const float* __restrict__ WP, const float* __restrict__ SE3pk,
    const float* __restrict__ jl, const float* __restrict__ trans,
    float* __restrict__ vout) {
  __shared__ float lds[8][16 * 17];
  int wid = threadIdx.x >> 5, lane = threadIdx.x & 31;
  int gw = blockIdx.x * 8 + wid;
  int b = gw / VTILE, vt = gw % VTILE;
  int v0 = vt * 16;
  int row = lane & 15, hi = lane >> 4, col = row;
  const float* S  = SE3pk + (size_t)b * 256;
  const float* Wr = WP + (size_t)(v0 + row) * 16;
  v8f acc = {0, 0, 0, 0, 0, 0, 0, 0};
#pragma unroll
  for (int ks = 0; ks < 4; ++ks) {
    v2f a  = ld2(Wr + ks * 4 + 2 * hi);
    v2f bb = ld2(S + ((ks * 2 + hi) * 16 + col) * 2);
    acc = wmma4(a, bb, acc);
  }
  // transpose SE3_v tile (rows = vertex, cols = xy) through LDS
#pragma unroll
  for (int i = 0; i < 8; ++i)
    lds[wid][(i + 8 * hi) * 17 + col] = acc[i];
  __syncthreads();
  if (lane < 16) {
    int v = v0 + lane;
    if (v < NV) {
      const float* m = &lds[wid][lane * 17];
      size_t base = (size_t)b * NVD + (size_t)v * 3;
      float x0 = vout[base], x1 = vout[base + 1], x2 = vout[base + 2];  // v_tpose (in place)
      float cx = jl[(size_t)b * 48 + 0], cy = jl[(size_t)b * 48 + 1], cz = jl[(size_t)b * 48 + 2];
      float tx = trans[b * 3 + 0], ty = trans[b * 3 + 1], tz = trans[b * 3 + 2];
      float r0 = m[0] * x0 + m[1] * x1 + m[2]  * x2 + m[3];
      float r1 = m[4] * x0 + m[5] * x1 + m[6]  * x2 + m[7];
      float r2 = m[8] * x0 + m[9] * x1 + m[10] * x2 + m[11];
      vout[base]     = r0 - cx + tx;
      vout[base + 1] = r1 - cy + ty;
      vout[base + 2] = r2 - cz + tz;
    }
  }
}

// ============================================================================
// K4: j_out gather/reorder. Tips read already-final v_out (center+trans folded).
// ============================================================================
__global__ __launch_bounds__(256) void k4_jout(
    const float* __restrict__ jl, const float* __restrict__ trans,
    const float* __restrict__ voutFinal, float* __restrict__ jout) {
  int idx = blockIdx.x * 256 + threadIdx.x;  // b*21 + k
  if (idx >= BS * 21) return;
  int b = idx / 21, k = idx % 21;
  int src = d_NEW_ORDER[k];
  float o0, o1, o2;
  if (src < 16) {
    const float* jb = jl + (size_t)b * 48;
    o0 = jb[3 * src + 0] - jb[0] + trans[b * 3 + 0];
    o1 = jb[3 * src + 1] - jb[1] + trans[b * 3 + 1];
    o2 = jb[3 * src + 2] - jb[2] + trans[b * 3 + 2];
  } else {
    size_t base = (size_t)b * NVD + (size_t)d_TIPS[src - 16] * 3;
    o0 = voutFinal[base]; o1 = voutFinal[base + 1]; o2 = voutFinal[base + 2];
  }
  jout[(size_t)idx * 3 + 0] = o0;
  jout[(size_t)idx * 3 + 1] = o1;
  jout[(size_t)idx * 3 + 2] = o2;
}

// ============================================================================
extern "C" void kernel_launch(void* const* d_in, const int* in_sizes, int n_in,
                              void* d_out, int out_size, void* d_ws, size_t ws_size,
                              hipStream_t stream) {
  const float* rootR     = (const float*)d_in[0];   // (BS,3,3)
  const float* pose      = (const float*)d_in[1];   // (BS,45)
  const float* shape     = (const float*)d_in[2];   // (BS,10)
  const float* trans     = (const float*)d_in[3];   // (BS,3)
  const float* HC        = (const float*)d_in[4];   // (45,45)
  const float* mean      = (const float*)d_in[5];   // (45,)
  const float* shapedirs = (const float*)d_in[6];   // (778,3,10)
  const float* posedirs  = (const float*)d_in[7];   // (778,3,135)
  const float* vtemp     = (const float*)d_in[8];   // (778,3)
  const float* Jreg      = (const float*)d_in[9];   // (16,778)
  const float* weights   = (const float*)d_in[10];  // (778,16)

  float* ws   = (float*)d_ws;
  float* vout = (float*)d_out;                    // (BS,778,3)
  float* jout = vout + (size_t)BS * NVD;          // (BS,21,3)

  k0_bpk<<<(KPAD * NSTR) / 256, 256, 0, stream>>>(posedirs, shapedirs, vtemp,
                                                  ws + OFF_BPK, ws + OFF_VTP);
  k0_js<<<3, 256, 0, stream>>>(Jreg, shapedirs, vtemp, ws + OFF_JS, ws + OFF_JT);
  {
    int packN = BS * 48 + 2304 + 48 + NVP * 16;
    k0_pack<<<(packN + 255) / 256, 256, 0, stream>>>(pose, HC, mean, weights,
                                                     ws + OFF_POSP, ws + OFF_HCP,
                                                     ws + OFF_MNP, ws + OFF_WP);
  }
  k_axis<<<96, 256, 0, stream>>>(ws + OFF_POSP, ws + OFF_HCP, ws + OFF_MNP,
                                 ws + OFF_AXIS);
  k1_prep<<<BS / 256, 256, 0, stream>>>(rootR, shape, ws + OFF_AXIS, ws + OFF_JS,
                                        ws + OFF_JT, ws + OFF_PF, ws + OFF_SE3,
                                        ws + OFF_JL);
  k2_vtpose<<<(NT_M * NT_N) / 8, 256, 0, stream>>>(ws + OFF_BPK, ws + OFF_PF,
                                                   ws + OFF_VTP, vout);
  k3_skin<<<(BS * VTILE) / 8, 256, 0, stream>>>(ws + OFF_WP, ws + OFF_SE3,
                                                ws + OFF_JL, trans, vout);
  k4_jout<<<(BS * 21 + 255) / 256, 256, 0, stream>>>(ws + OFF_JL, trans, vout, jout);
}